// OptNet_40724879901090
// MI455X (gfx1250) — compile-verified
//
#include <hip/hip_runtime.h>

// Batched primal-dual IPM for  min 0.5 z'Qz + p'z  s.t. z >= 0
// with Q = 0.1*I, G = -I, h = 0.  The reduced KKT matrix
// K = Q + G' diag(lam/s) G = diag(0.1 + lam/s) is DIAGONAL, so the
// reference's batched dense solve is exactly an elementwise divide.
// One thread = one QP problem; all state lives in VGPRs.

#define NXV   24
#define NITER 20
#define QPEN  0.1f
#define CSIG  0.1f
#define BIGF  1e9f

constexpr int BPB = 128;  // threads (= batch problems) per block

__global__ __launch_bounds__(BPB)
void optnet_ipm_kernel(const float* __restrict__ puzzles,
                       float* __restrict__ out, int B) {
    const int b = blockIdx.x * BPB + threadIdx.x;
    if (b >= B) return;

    const float* __restrict__ src = puzzles + (size_t)b * NXV;
    float* __restrict__ dst = out + (size_t)b * NXV;

    // gfx1250 prefetch path (global_prefetch_b8): pull our 96B row toward L2/L0
    __builtin_prefetch(src, 0, 0);

    // Load p = -puzzles row via 6x 128-bit loads (row is 96B, 16B-aligned).
    float p[NXV], z[NXV], s[NXV], lam[NXV], dz[NXV];
    const float4* src4 = reinterpret_cast<const float4*>(src);
    #pragma unroll
    for (int q = 0; q < NXV / 4; ++q) {
        float4 v = src4[q];
        p[4 * q + 0] = -v.x;
        p[4 * q + 1] = -v.y;
        p[4 * q + 2] = -v.z;
        p[4 * q + 3] = -v.w;
    }
    #pragma unroll
    for (int j = 0; j < NXV; ++j) {
        z[j] = 0.0f;
        s[j] = 1.0f;
        lam[j] = 1.0f;
    }

    #pragma unroll 1
    for (int it = 0; it < NITER; ++it) {
        // mu = SIGMA * mean(s .* lam)
        float acc = 0.0f;
        #pragma unroll
        for (int j = 0; j < NXV; ++j) acc += s[j] * lam[j];
        const float mu = CSIG * acc * (1.0f / (float)NXV);

        // Pass 1: Newton direction (diagonal K) + fraction-to-boundary min.
        float minr = BIGF;
        #pragma unroll
        for (int j = 0; j < NXV; ++j) {
            const float sj = s[j], lj = lam[j], zj = z[j];
            const float inv_s = 1.0f / sj;
            const float rd = QPEN * zj + p[j] - lj;   // r_d = Qz + p + G'lam
            const float rp = sj - zj;                  // r_p = Gz + s - h
            const float rc = sj * lj - mu;             // r_c = s.*lam - mu
            const float w  = lj * inv_s;               // lam/s
            // rhs = -(r_d + G'((lam*r_p - r_c)/s)),  G' = -I
            const float rhs = -(rd - (lj * rp - rc) * inv_s);
            const float d   = rhs / (QPEN + w);        // dz = K^-1 rhs (diag)
            dz[j] = d;
            const float dsj = d - rp;                  // ds = -r_p - G dz
            const float dlj = (-rc - lj * dsj) * inv_s;
            const float rs  = (dsj < 0.0f) ? (-sj / dsj) : BIGF;
            const float rl  = (dlj < 0.0f) ? (-lj / dlj) : BIGF;
            minr = fminf(minr, fminf(rs, rl));
        }
        const float alpha = fminf(1.0f, 0.99f * minr);

        // Pass 2: recompute ds/dlam (cheap) and take the step.
        #pragma unroll
        for (int j = 0; j < NXV; ++j) {
            const float sj = s[j], lj = lam[j];
            const float rp  = sj - z[j];
            const float rc  = sj * lj - mu;
            const float dsj = dz[j] - rp;
            const float dlj = (-rc - lj * dsj) / sj;
            z[j]   += alpha * dz[j];
            s[j]   += alpha * dsj;
            lam[j] += alpha * dlj;
        }
    }

    // Store z via 6x 128-bit stores.
    float4* dst4 = reinterpret_cast<float4*>(dst);
    #pragma unroll
    for (int q = 0; q < NXV / 4; ++q) {
        float4 v;
        v.x = z[4 * q + 0];
        v.y = z[4 * q + 1];
        v.z = z[4 * q + 2];
        v.w = z[4 * q + 3];
        dst4[q] = v;
    }
}

extern "C" void kernel_launch(void* const* d_in, const int* in_sizes, int n_in,
                              void* d_out, int out_size, void* d_ws, size_t ws_size,
                              hipStream_t stream) {
    const float* puzzles = (const float*)d_in[0];
    float* out = (float*)d_out;
    const int total = in_sizes[0];      // 32768 * 12 * 2
    const int B = total / NXV;          // 32768 QP problems
    const int blocks = (B + BPB - 1) / BPB;
    optnet_ipm_kernel<<<blocks, BPB, 0, stream>>>(puzzles, out, B);
}